// FNet_Model_PureTorch_13400297964048
// MI455X (gfx1250) — compile-verified
//
#include <hip/hip_runtime.h>
#include <hip/hip_bf16.h>
#include <math.h>

typedef __attribute__((ext_vector_type(2))) float v2f;
typedef __attribute__((ext_vector_type(8))) float v8f;

#define DMODEL 768
#define DFF    3072
#define NSEQ   2048
#define NBATCH 8
#define NLAYER 12
#define NVOCAB 32000

// ---------------------------------------------------------------------------
// Embedding gather: one block per token, 768 floats per row.
// ---------------------------------------------------------------------------
__global__ __launch_bounds__(256) void embed_kernel(
    const int* __restrict__ tokens, const float* __restrict__ emb,
    float* __restrict__ x)
{
  int t = blockIdx.x;
  int tok = tokens[t];
  const float* src = emb + (size_t)tok * DMODEL;
  float* dst = x + (size_t)t * DMODEL;
#pragma unroll
  for (int i = 0; i < 3; ++i) {
    int c = threadIdx.x + i * 256;
    dst[c] = src[c];
  }
}

// ---------------------------------------------------------------------------
// LayerNorm over last dim (768). One block per row.
// ---------------------------------------------------------------------------
__global__ __launch_bounds__(256) void layernorm_kernel(
    const float* __restrict__ x, const float* __restrict__ g,
    const float* __restrict__ b, float* __restrict__ out)
{
  int row = blockIdx.x;
  const float* xr = x + (size_t)row * DMODEL;
  float loc[3];
  float s = 0.f, s2 = 0.f;
#pragma unroll
  for (int i = 0; i < 3; ++i) {
    float v = xr[threadIdx.x + i * 256];
    loc[i] = v; s += v; s2 += v * v;
  }
  __shared__ float sh1[256], sh2[256];
  sh1[threadIdx.x] = s; sh2[threadIdx.x] = s2;
  __syncthreads();
  for (int off = 128; off > 0; off >>= 1) {
    if (threadIdx.x < off) {
      sh1[threadIdx.x] += sh1[threadIdx.x + off];
      sh2[threadIdx.x] += sh2[threadIdx.x + off];
    }
    __syncthreads();
  }
  float mean = sh1[0] * (1.f / DMODEL);
  float var  = sh2[0] * (1.f / DMODEL) - mean * mean;
  float inv  = rsqrtf(var + 1e-5f);
  float* orow = out + (size_t)row * DMODEL;
#pragma unroll
  for (int i = 0; i < 3; ++i) {
    int c = threadIdx.x + i * 256;
    orow[c] = (loc[i] - mean) * inv * g[c] + b[c];
  }
}

// ---------------------------------------------------------------------------
// DFT matrix generation (fp64 sincos, ang = -2*pi*(k*j mod N)/N):
//   cosM = cos(ang), sinM = sinSign * sin(ang)
// ---------------------------------------------------------------------------
__global__ __launch_bounds__(256) void fill_dft_kernel(
    float* __restrict__ cosM, float* __restrict__ sinM, int N, float sinSign)
{
  long long i = (long long)blockIdx.x * 256 + threadIdx.x;
  long long total = (long long)N * N;
  if (i >= total) return;
  int k = (int)(i / N), j = (int)(i % N);
  long long m = ((long long)k * (long long)j) % N;
  double ang = -2.0 * 3.14159265358979323846 * (double)m / (double)N;
  cosM[i] = (float)cos(ang);
  sinM[i] = sinSign * (float)sin(ang);
}

// ---------------------------------------------------------------------------
// Generic strided-batched fp32 WMMA GEMM:
//   C[b] (+)= act( A[b] * B[b] + bias )
// A: row-major [M,K] with row stride lda, B: row-major [K,N], C: row-major [M,N].
// Requirements: K % 16 == 0, N % 128 == 0. M may be arbitrary (row-clamped
// loads + row-masked stores). Block tile 128x128, 8 waves (4 M x 2 N), each
// wave = 2x4 WMMA 16x16x4 f32 accumulators, BK = 16, double-buffered LDS.
// Both A and B(t) are stored so that each fragment is one aligned ds_load_b64.
// ---------------------------------------------------------------------------
#define BM 128
#define BN 128
#define BK 16
#define AS_STRIDE 18    // 16 + 2 pad: 8B-aligned b64 reads, conflict-free
#define BT_STRIDE 18    // B transposed: Bs[BN][BK+2]

__device__ __forceinline__ float gelu_exact(float v) {
  return 0.5f * v * (1.f + erff(v * 0.70710678118654752f));
}

__device__ __forceinline__ void g_load_tiles(
    const float* __restrict__ A, const float* __restrict__ B,
    int tid, int tileM, int tileN, int M, int N, int lda, int k0,
    float* aReg, float* bReg)
{
#pragma unroll
  for (int i = 0; i < 8; ++i) {                 // A tile: 128x16
    int idx = tid + i * 256;
    int r = idx >> 4, c = idx & 15;
    int gr = tileM + r;
    gr = (gr < M) ? gr : (M - 1);               // clamp: OOB rows feed only
    aReg[i] = A[(size_t)gr * lda + (k0 + c)];   // OOB C rows (never stored)
  }
#pragma unroll
  for (int i = 0; i < 8; ++i) {                 // B tile: 16x128
    int idx = tid + i * 256;
    int r = idx >> 7, c = idx & 127;
    bReg[i] = B[(size_t)(k0 + r) * N + (tileN + c)];
  }
}

__device__ __forceinline__ void lds_store_tiles(
    float* __restrict__ as, float* __restrict__ bs, int tid,
    const float* aReg, const float* bReg)
{
#pragma unroll
  for (int i = 0; i < 8; ++i) {
    int idx = tid + i * 256;
    as[(idx >> 4) * AS_STRIDE + (idx & 15)] = aReg[i];
  }
#pragma unroll
  for (int i = 0; i < 8; ++i) {                 // transpose B: Bs[n][k]
    int idx = tid + i * 256;
    bs[(idx & 127) * BT_STRIDE + (idx >> 7)] = bReg[i];
  }
}

__global__ __launch_bounds__(256) void gemm_wmma_f32_kernel(
    const float* __restrict__ A, const float* __restrict__ B,
    float* __restrict__ C, const float* __restrict__ bias,
    int M, int N, int K, int lda,
    long long strideA, long long strideB, long long strideC,
    int addC, int act)
{
  __shared__ float As[2][BM * AS_STRIDE];
  __shared__ float Bs[2][BN * BT_STRIDE];

  int bz = blockIdx.z;
  A += (size_t)bz * strideA;
  B += (size_t)bz * strideB;
  C += (size_t)bz * strideC;

  int tileM = blockIdx.y * BM;
  int tileN = blockIdx.x * BN;
  int tid  = threadIdx.x;
  int lane = tid & 31;
  int wave = tid >> 5;
  int wm = wave & 3;        // 4 waves along M (32 rows each)
  int wn = wave >> 2;       // 2 waves along N (64 cols each)
  int ln16 = lane & 15;
  int half = lane >> 4;

  v8f acc[2][4];
#pragma unroll
  for (int t = 0; t < 2; ++t)
#pragma unroll
    for (int u = 0; u < 4; ++u)
      acc[t][u] = (v8f)0.f;

  int KT = K / BK;
  float aReg[8], bReg[8];

  // Preload first K-chunk.
  g_load_tiles(A, B, tid, tileM, tileN, M, N, lda, 0, aReg, bReg);
  lds_store_tiles(As[0], Bs[0], tid, aReg, bReg);
  __syncthreads();

  for (int kt = 0; kt < KT; ++kt) {
    int cur = kt & 1;
    if (kt + 1 < KT)
      g_load_tiles(A, B, tid, tileM, tileN, M, N, lda, (kt + 1) * BK,
                   aReg, bReg);
    if (kt + 2 < KT) {
      // Hint the K-chunk two iterations out into cache (global_prefetch_b8).
      int k2 = (kt + 2) * BK;
      int ar = tileM + (tid >> 1);
      ar = (ar < M) ? ar : (M - 1);
      __builtin_prefetch(&A[(size_t)ar * lda + k2 + (tid & 1) * 8], 0, 3);
      __builtin_prefetch(&B[(size_t)(k2 + (tid >> 4)) * N +
                            tileN + (tid & 15) * 8], 0, 3);
    }

    const float* as = As[cur];
    const float* bs = Bs[cur];
#pragma unroll
    for (int kk = 0; kk < BK; kk += 4) {
      // 16x4 f32 fragments: lanes 0-15 hold K=kk,kk+1; lanes 16-31 K=kk+2,kk+3
      int krow = kk + (half << 1);
      v2f afr[2], bfr[4];
#pragma unroll
      for (int t = 0; t < 2; ++t) {
        const float* p = &as[(wm * 32 + t * 16 + ln16) * AS_STRIDE + krow];
        afr[t] = *(const v2f*)p;   // aligned ds_load_b64
      }
#pragma unroll
      for (int u = 0; u < 4; ++u) {
        const float* p = &bs[(wn * 64 + u * 16 + ln16) * BT_STRIDE + krow];
        bfr[u] = *(const v2f*)p;   // aligned ds_load_b64 (B transposed in LDS)
      }
#pragma unroll
      for (int t = 0; t < 2; ++t)
#pragma unroll
        for (int u = 0; u < 4; ++u)
          acc[t][u] = __builtin_amdgcn_wmma_f32_16x16x4_f32(
              false, afr[t], false, bfr[u], (short)0, acc[t][u],
              false, false);
    }

    if (kt + 1 < KT)
      lds_store_tiles(As[cur ^ 1], Bs[cur ^ 1], tid, aReg, bReg);
    __syncthreads();
  }

  // Epilogue: bias -> activation -> optional C accumulate -> store.
#pragma unroll
  for (int t = 0; t < 2; ++t) {
#pragma unroll
    for (int u = 0; u < 4; ++u) {
      int col = tileN + wn * 64 + u * 16 + ln16;
      float bv = bias ? bias[col] : 0.f;
#pragma unroll
      for (int r = 0; r < 8; ++r) {
        int row = tileM + wm * 32 + t * 16 + half * 8 + r;
        if (row < M) {
          float v = acc[t][u][r] + bv;
          if (act) v = gelu_exact(v);
          size_t o = (size_t)row * N + col;
          if (addC) v += C[o];
          C[o] = v;
        }
      }
    }
  }
}

// ---------------------------------------------------------------------------
// Host-side orchestration.
// ---------------------------------------------------------------------------
static inline void launch_gemm(const float* A, const float* B, float* C,
                               const float* bias, int M, int N, int K, int lda,
                               long long sA, long long sB, long long sC,
                               int batch, int addC, int act, hipStream_t stream)
{
  dim3 grid(N / BN, (M + BM - 1) / BM, batch);
  gemm_wmma_f32_kernel<<<grid, 256, 0, stream>>>(A, B, C, bias, M, N, K, lda,
                                                 sA, sB, sC, addC, act);
}

extern "C" void kernel_launch(void* const* d_in, const int* in_sizes, int n_in,
                              void* d_out, int out_size, void* d_ws, size_t ws_size,
                              hipStream_t stream) {
  const int*   tokens = (const int*)d_in[0];
  const float* emb    = (const float*)d_in[1];
  const float* ln1_g  = (const float*)d_in[2];
  const float* ln1_b  = (const float*)d_in[3];
  const float* ln2_g  = (const float*)d_in[4];
  const float* ln2_b  = (const float*)d_in[5];
  const float* W1     = (const float*)d_in[6];
  const float* b1     = (const float*)d_in[7];
  const float* W2     = (const float*)d_in[8];
  const float* b2     = (const float*)d_in[9];
  const float* Wout   = (const float*)d_in[10];
  const float* bout   = (const float*)d_in[11];
  float* out = (float*)d_out;

  const int B = NBATCH, S = NSEQ, D = DMODEL, F = DFF, L = NLAYER, V = NVOCAB;
  const int T = B * S;

  float* ws = (float*)d_ws;
  size_t off = 0;
  float* x    = ws + off; off += (size_t)T * D;
  float* h    = ws + off; off += (size_t)T * D;
  float* re   = ws + off; off += (size_t)T * D;
  float* im   = ws + off; off += (size_t)T * D;
  float* ff   = ws + off; off += (size_t)T * F;
  float* cosD = ws + off; off += (size_t)D * D;
  float* sinD = ws + off; off += (size_t)D * D;
  float* cosS = ws + off; off += (size_t)S * S;
  float* sinS = ws + off; off += (size_t)S * S;
  if (ws_size < off * sizeof(float)) return;  // scratch too small: bail out

  // DFT coefficient matrices (fp64 sincos for exact twiddles, once per call).
  // D-axis forward DFT: Re-coeff = cos(ang), Im-coeff = sin(ang), ang=-2*pi*kj/N.
  fill_dft_kernel<<<(D * D + 255) / 256, 256, 0, stream>>>(cosD, sinD, D, 1.f);
  // S-axis: Re(out) = cos(2*pi*ts/S)*re + sin(2*pi*ts/S)*im -> flip sin sign.
  fill_dft_kernel<<<(S * S + 255) / 256, 256, 0, stream>>>(cosS, sinS, S, -1.f);

  // x = embed[tokens]
  embed_kernel<<<T, 256, 0, stream>>>(tokens, emb, x);

  for (int l = 0; l < L; ++l) {
    // h = LN1(x)
    layernorm_kernel<<<T, 256, 0, stream>>>(x, ln1_g + (size_t)l * D,
                                            ln1_b + (size_t)l * D, h);
    // D-axis DFT as GEMM: re = h @ cosD ; im = h @ sinD
    launch_gemm(h, cosD, re, nullptr, T, D, D, D, 0, 0, 0, 1, 0, 0, stream);
    launch_gemm(h, sinD, im, nullptr, T, D, D, D, 0, 0, 0, 1, 0, 0, stream);
    // S-axis DFT fused with residual: x += cosS @ re[b] + sinS @ im[b]
    launch_gemm(cosS, re, x, nullptr, S, D, S, S,
                0, (long long)S * D, (long long)S * D, B, 1, 0, stream);
    launch_gemm(sinS, im, x, nullptr, S, D, S, S,
                0, (long long)S * D, (long long)S * D, B, 1, 0, stream);
    // h = LN2(x)
    layernorm_kernel<<<T, 256, 0, stream>>>(x, ln2_g + (size_t)l * D,
                                            ln2_b + (size_t)l * D, h);
    // ff = gelu(h @ W1_l + b1_l)
    launch_gemm(h, W1 + (size_t)l * D * F, ff, b1 + (size_t)l * F,
                T, F, D, D, 0, 0, 0, 1, 0, 1, stream);
    // x += ff @ W2_l + b2_l
    launch_gemm(ff, W2 + (size_t)l * F * D, x, b2 + (size_t)l * D,
                T, D, F, F, 0, 0, 0, 1, 1, 0, stream);
  }

  // Logits from last position: out[b,:] = x[b, S-1, :] @ Wout + bout
  // A rows are strided by S*D (one row per batch element).
  launch_gemm(x + (size_t)(S - 1) * D, Wout, out, bout,
              B, V, D, S * D, 0, 0, 0, 1, 0, 0, stream);
}